// MultiResFeatureGrid2D_8933531976487
// MI455X (gfx1250) — compile-verified
//
#include <hip/hip_runtime.h>

#define BLOCK 256
#define PPT 8   // points per thread

// LDS staging for levels 0..2 (16^2, 32^2, 64^2): 256+1024+4096 = 5376 float2 = 43008 B
#define L0_OFF 0
#define L1_OFF 256
#define L2_OFF 1280
#define LDS_F2 5376

#if defined(__AMDGCN__) && __has_builtin(__builtin_amdgcn_global_load_async_to_lds_b128)
#define HAVE_ASYNC_LDS 1
#else
#define HAVE_ASYNC_LDS 0
#endif

#if HAVE_ASYNC_LDS
typedef int v4i __attribute__((ext_vector_type(4)));
typedef __attribute__((address_space(1))) v4i* gptr_v4i;  // global
typedef __attribute__((address_space(3))) v4i* lptr_v4i;  // LDS
#endif

typedef float vf2 __attribute__((ext_vector_type(2)));  // native vector for NT ld/st

__device__ __forceinline__ float2 lerp2(float2 a, float2 b, float t) {
    float2 r;
    r.x = fmaf(t, b.x - a.x, a.x);
    r.y = fmaf(t, b.y - a.y, a.y);
    return r;
}

// Non-temporal streaming store of a float2 (write-once output; keep L2 for grids).
__device__ __forceinline__ void st_nt(float2* p, float2 v) {
    vf2 t; t.x = v.x; t.y = v.y;
    __builtin_nontemporal_store(t, (vf2*)p);
}

// Non-temporal streaming load of a float2 (read-once coords).
__device__ __forceinline__ float2 ld_nt(const float2* p) {
    vf2 t = __builtin_nontemporal_load((const vf2*)p);
    float2 r; r.x = t.x; r.y = t.y;
    return r;
}

// Bilinear interp of a 2-channel RxR grid at (x,y) in [0,1).
// Works for both global and LDS-resident grids (inlined; address space inferred).
template <int R>
__device__ __forceinline__ float2 interp(const float2* __restrict__ g, float x, float y) {
    float xs = x * (float)(R - 1);
    float ys = y * (float)(R - 1);
    int x0 = min((int)xs, R - 2);   // xs >= 0, so (int) == floor
    int y0 = min((int)ys, R - 2);
    float fx = xs - (float)x0;
    float fy = ys - (float)y0;
    int i00 = x0 + y0 * R;
    float2 f00 = g[i00];
    float2 f10 = g[i00 + 1];
    float2 f01 = g[i00 + R];
    float2 f11 = g[i00 + R + 1];
    return lerp2(lerp2(f00, f10, fx), lerp2(f01, f11, fx), fy);
}

// Cooperative copy of n16 x 16-byte chunks from global to LDS.
// Uses CDNA5 async global->LDS DMA (ASYNCcnt, global_load_async_to_lds_b128).
__device__ __forceinline__ void stage16(const float4* __restrict__ src, float4* dst, int n16) {
    for (int j = threadIdx.x; j < n16; j += BLOCK) {
#if HAVE_ASYNC_LDS
        __builtin_amdgcn_global_load_async_to_lds_b128(
            (gptr_v4i)(src + j),
            (lptr_v4i)(dst + j),
            0, 0);
#else
        dst[j] = src[j];
#endif
    }
}

__global__ __launch_bounds__(BLOCK)
void mrfg_kernel(const float2* __restrict__ coords,
                 const float2* __restrict__ g0, const float2* __restrict__ g1,
                 const float2* __restrict__ g2, const float2* __restrict__ g3,
                 const float2* __restrict__ g4, const float2* __restrict__ g5,
                 const float2* __restrict__ g6,
                 float2* __restrict__ out, int n) {
    __shared__ __align__(16) float2 sg[LDS_F2];

    stage16((const float4*)g0, (float4*)(sg + L0_OFF), 128);
    stage16((const float4*)g1, (float4*)(sg + L1_OFF), 512);
    stage16((const float4*)g2, (float4*)(sg + L2_OFF), 2048);

#if HAVE_ASYNC_LDS
#if __has_builtin(__builtin_amdgcn_s_wait_asynccnt)
    __builtin_amdgcn_s_wait_asynccnt(0);
#else
    asm volatile("s_wait_asynccnt 0" ::: "memory");
#endif
#endif
    __syncthreads();

    const float HI = 1.0f - 1e-6f;
    int base = blockIdx.x * (BLOCK * PPT) + threadIdx.x;

#pragma unroll
    for (int k = 0; k < PPT; ++k) {
        int i = base + k * BLOCK;
        if (i < n) {
            float2 c = ld_nt(coords + i);
            float x = fminf(fmaxf(c.x, 0.0f), HI);
            float y = fminf(fmaxf(c.y, 0.0f), HI);

            float2 r0 = interp<16>(sg + L0_OFF, x, y);     // LDS
            float2 r1 = interp<32>(sg + L1_OFF, x, y);     // LDS
            float2 r2 = interp<64>(sg + L2_OFF, x, y);     // LDS
            float2 r3 = interp<128>(g3, x, y);             // L2-resident
            float2 r4 = interp<256>(g4, x, y);
            float2 r5 = interp<512>(g5, x, y);
            float2 r6 = interp<1024>(g6, x, y);

            float2* o = out + (size_t)i * 7;               // 14 floats = 7 float2, 8B aligned
            st_nt(o + 0, r0); st_nt(o + 1, r1); st_nt(o + 2, r2); st_nt(o + 3, r3);
            st_nt(o + 4, r4); st_nt(o + 5, r5); st_nt(o + 6, r6);
        }
    }
}

extern "C" void kernel_launch(void* const* d_in, const int* in_sizes, int n_in,
                              void* d_out, int out_size, void* d_ws, size_t ws_size,
                              hipStream_t stream) {
    const float2* coords = (const float2*)d_in[0];
    const float2* g0 = (const float2*)d_in[1];
    const float2* g1 = (const float2*)d_in[2];
    const float2* g2 = (const float2*)d_in[3];
    const float2* g3 = (const float2*)d_in[4];
    const float2* g4 = (const float2*)d_in[5];
    const float2* g5 = (const float2*)d_in[6];
    const float2* g6 = (const float2*)d_in[7];
    float2* out = (float2*)d_out;

    int n = in_sizes[0] / 2;  // number of points
    int blocks = (n + BLOCK * PPT - 1) / (BLOCK * PPT);
    mrfg_kernel<<<blocks, BLOCK, 0, stream>>>(coords, g0, g1, g2, g3, g4, g5, g6, out, n);
}